// GSModel_68779606278856
// MI455X (gfx1250) — compile-verified
//
#include <hip/hip_runtime.h>
#include <math.h>

#define GS_N   8192
#define GS_H   256
#define GS_W   256
#define GS_HW  (GS_H * GS_W)

typedef float v2f __attribute__((ext_vector_type(2)));
typedef float v8f __attribute__((ext_vector_type(8)));

__device__ __forceinline__ float sigmoidf(float v) { return 1.0f / (1.0f + __expf(-v)); }

// ---------------------------------------------------------------------------
// Phase 1: per-gaussian preprocessing. One thread per gaussian.
// ---------------------------------------------------------------------------
__global__ void gs_prep(const float* __restrict__ pws, const float* __restrict__ shs,
                        const float* __restrict__ araw, const float* __restrict__ sraw,
                        const float* __restrict__ rraw,
                        const float* __restrict__ Rcw, const float* __restrict__ tcw,
                        const float* __restrict__ intr,
                        float* __restrict__ depth_o, float* __restrict__ ux_o,
                        float* __restrict__ uy_o, float* __restrict__ ca_o,
                        float* __restrict__ cb_o, float* __restrict__ cc_o,
                        float* __restrict__ al_o, float* __restrict__ col_o,
                        float* __restrict__ areas_o, int N, int H, int W) {
  int i = blockIdx.x * blockDim.x + threadIdx.x;
  if (i >= N) return;

  const float fx = intr[0], fy = intr[1], cx = intr[2], cy = intr[3];
  const float R00 = Rcw[0], R01 = Rcw[1], R02 = Rcw[2];
  const float R10 = Rcw[3], R11 = Rcw[4], R12 = Rcw[5];
  const float R20 = Rcw[6], R21 = Rcw[7], R22 = Rcw[8];
  const float t0 = tcw[0], t1 = tcw[1], t2 = tcw[2];

  const float pwx = pws[i * 3 + 0], pwy = pws[i * 3 + 1], pwz = pws[i * 3 + 2];

  float alpha = sigmoidf(araw[i]);
  float s0 = __expf(sraw[i * 3 + 0]);
  float s1 = __expf(sraw[i * 3 + 1]);
  float s2 = __expf(sraw[i * 3 + 2]);

  float q0 = rraw[i * 4 + 0], q1 = rraw[i * 4 + 1], q2 = rraw[i * 4 + 2], q3 = rraw[i * 4 + 3];
  float qn = rsqrtf(q0 * q0 + q1 * q1 + q2 * q2 + q3 * q3);
  float qw = q0 * qn, qx = q1 * qn, qy = q2 * qn, qz = q3 * qn;

  // camera-space point: pcs = Rcw @ pw + tcw
  float pcx = R00 * pwx + R01 * pwy + R02 * pwz + t0;
  float pcy = R10 * pwx + R11 * pwy + R12 * pwz + t1;
  float pcz = R20 * pwx + R21 * pwy + R22 * pwz + t2;
  float depth = pcz;
  float zs = (depth > 0.2f) ? depth : 1.0f;
  float inv_z = 1.0f / zs;

  float ux = fx * pcx * inv_z + cx;
  float uy = fy * pcy * inv_z + cy;

  // rotation matrix
  float r00 = 1.0f - 2.0f * (qy * qy + qz * qz);
  float r01 = 2.0f * (qx * qy - qw * qz);
  float r02 = 2.0f * (qx * qz + qw * qy);
  float r10 = 2.0f * (qx * qy + qw * qz);
  float r11 = 1.0f - 2.0f * (qx * qx + qz * qz);
  float r12 = 2.0f * (qy * qz - qw * qx);
  float r20 = 2.0f * (qx * qz - qw * qy);
  float r21 = 2.0f * (qy * qz + qw * qx);
  float r22 = 1.0f - 2.0f * (qx * qx + qy * qy);

  // M = R * diag(s), cov3d = M M^T
  float m00 = r00 * s0, m01 = r01 * s1, m02 = r02 * s2;
  float m10 = r10 * s0, m11 = r11 * s1, m12 = r12 * s2;
  float m20 = r20 * s0, m21 = r21 * s1, m22 = r22 * s2;
  float c00 = m00 * m00 + m01 * m01 + m02 * m02;
  float c01 = m00 * m10 + m01 * m11 + m02 * m12;
  float c02 = m00 * m20 + m01 * m21 + m02 * m22;
  float c11 = m10 * m10 + m11 * m11 + m12 * m12;
  float c12 = m10 * m20 + m11 * m21 + m12 * m22;
  float c22 = m20 * m20 + m21 * m21 + m22 * m22;

  float tanfx = (float)W / (2.0f * fx);
  float tanfy = (float)H / (2.0f * fy);
  float tx = fminf(fmaxf(pcx * inv_z, -1.3f * tanfx), 1.3f * tanfx) * zs;
  float ty = fminf(fmaxf(pcy * inv_z, -1.3f * tanfy), 1.3f * tanfy) * zs;

  float J00 = fx * inv_z, J02 = -fx * tx * inv_z * inv_z;
  float J11 = fy * inv_z, J12 = -fy * ty * inv_z * inv_z;

  // T2 = J @ Rcw  (2x3)
  float T00 = J00 * R00 + J02 * R20;
  float T01 = J00 * R01 + J02 * R21;
  float T02 = J00 * R02 + J02 * R22;
  float T10 = J11 * R10 + J12 * R20;
  float T11 = J11 * R11 + J12 * R21;
  float T12 = J11 * R12 + J12 * R22;

  // cov2d = T2 cov3d T2^T + 0.3 I
  float a0 = c00 * T00 + c01 * T01 + c02 * T02;
  float a1 = c01 * T00 + c11 * T01 + c12 * T02;
  float a2 = c02 * T00 + c12 * T01 + c22 * T02;
  float b0 = c00 * T10 + c01 * T11 + c02 * T12;
  float b1 = c01 * T10 + c11 * T11 + c12 * T12;
  float b2 = c02 * T10 + c12 * T11 + c22 * T12;
  float A_ = T00 * a0 + T01 * a1 + T02 * a2 + 0.3f;
  float B_ = T10 * a0 + T11 * a1 + T12 * a2;
  float C_ = T10 * b0 + T11 * b1 + T12 * b2 + 0.3f;

  float det = A_ * C_ - B_ * B_;
  bool valid = (depth > 0.2f) && (det > 0.0f);
  float det_s = valid ? det : 1.0f;
  float inv_det = 1.0f / det_s;

  float mid = 0.5f * (A_ + C_);
  float lam = mid + sqrtf(fmaxf(mid * mid - det, 0.1f));
  float radius = valid ? ceilf(3.0f * sqrtf(lam)) : 0.0f;
  areas_o[i * 2 + 0] = radius;
  areas_o[i * 2 + 1] = radius;

  // view direction: twc = -Rcw^T tcw
  float twx = -(R00 * t0 + R10 * t1 + R20 * t2);
  float twy = -(R01 * t0 + R11 * t1 + R21 * t2);
  float twz = -(R02 * t0 + R12 * t1 + R22 * t2);
  float dx = pwx - twx, dy = pwy - twy, dz = pwz - twz;
  float dn = rsqrtf(dx * dx + dy * dy + dz * dz);
  float x = dx * dn, y = dy * dn, z = dz * dn;
  float xx = x * x, yy = y * y, zz = z * z;
  float xy = x * y, yz = y * z, xz = x * z;

  const float C0 = 0.28209479177387814f;
  const float C1 = 0.4886025119029199f;
  const float C2_0 = 1.0925484305920792f, C2_1 = -1.0925484305920792f;
  const float C2_2 = 0.31539156525252005f, C2_3 = -1.0925484305920792f, C2_4 = 0.5462742152960396f;
  const float C3_0 = -0.5900435899266435f, C3_1 = 2.890611442640554f, C3_2 = -0.4570457994644658f;
  const float C3_3 = 0.3731763325901154f, C3_4 = -0.4570457994644658f, C3_5 = 1.445305721320277f;
  const float C3_6 = -0.5900435899266435f;

  const float* sh = shs + (size_t)i * 48;
#pragma unroll
  for (int c = 0; c < 3; ++c) {
    float r = C0 * sh[0 * 3 + c];
    r += -C1 * y * sh[1 * 3 + c] + C1 * z * sh[2 * 3 + c] - C1 * x * sh[3 * 3 + c];
    r += C2_0 * xy * sh[4 * 3 + c] + C2_1 * yz * sh[5 * 3 + c] +
         C2_2 * (2.0f * zz - xx - yy) * sh[6 * 3 + c] + C2_3 * xz * sh[7 * 3 + c] +
         C2_4 * (xx - yy) * sh[8 * 3 + c];
    r += C3_0 * y * (3.0f * xx - yy) * sh[9 * 3 + c] + C3_1 * xy * z * sh[10 * 3 + c] +
         C3_2 * y * (4.0f * zz - xx - yy) * sh[11 * 3 + c] +
         C3_3 * z * (2.0f * zz - 3.0f * xx - 3.0f * yy) * sh[12 * 3 + c] +
         C3_4 * x * (4.0f * zz - xx - yy) * sh[13 * 3 + c] +
         C3_5 * z * (xx - yy) * sh[14 * 3 + c] + C3_6 * x * (xx - 3.0f * yy) * sh[15 * 3 + c];
    col_o[i * 3 + c] = fmaxf(r + 0.5f, 0.0f);
  }

  depth_o[i] = depth;
  ux_o[i] = ux;
  uy_o[i] = uy;
  ca_o[i] = C_ * inv_det;   // cinv[0] = c/det
  cb_o[i] = -B_ * inv_det;  // cinv[1] = -b/det
  cc_o[i] = A_ * inv_det;   // cinv[2] = a/det
  al_o[i] = valid ? alpha : 0.0f;  // fold validity into alpha (gates alp to 0)
}

// ---------------------------------------------------------------------------
// Phase 2: single-workgroup bitonic argsort by depth (ascending). N = 8192.
// ---------------------------------------------------------------------------
__global__ __launch_bounds__(1024) void gs_sort(const float* __restrict__ depth,
                                                int* __restrict__ order, int n) {
  __shared__ float key[GS_N];
  __shared__ int idx[GS_N];
  int tid = threadIdx.x;
  for (int i = tid; i < n; i += 1024) { key[i] = depth[i]; idx[i] = i; }
  __syncthreads();
  for (int k = 2; k <= n; k <<= 1) {
    for (int j = k >> 1; j > 0; j >>= 1) {
      for (int i = tid; i < n; i += 1024) {
        int l = i ^ j;
        if (l > i) {
          bool asc = ((i & k) == 0);
          float ki = key[i], kl = key[l];
          bool sw = asc ? (ki > kl) : (ki < kl);
          if (sw) {
            key[i] = kl; key[l] = ki;
            int ti = idx[i]; idx[i] = idx[l]; idx[l] = ti;
          }
        }
      }
      __syncthreads();
    }
  }
  for (int i = tid; i < n; i += 1024) order[i] = idx[i];
}

// ---------------------------------------------------------------------------
// Phase 3a: gather sorted per-gaussian params into packed struct (8 floats).
// ---------------------------------------------------------------------------
__global__ void gs_gather(const int* __restrict__ order,
                          const float* __restrict__ ux, const float* __restrict__ uy,
                          const float* __restrict__ ca, const float* __restrict__ cb,
                          const float* __restrict__ cc, const float* __restrict__ al,
                          float* __restrict__ g8, int N) {
  int i = blockIdx.x * blockDim.x + threadIdx.x;
  if (i >= N) return;
  int s = order[i];
  g8[i * 8 + 0] = ux[s];
  g8[i * 8 + 1] = uy[s];
  g8[i * 8 + 2] = ca[s];
  g8[i * 8 + 3] = cb[s];
  g8[i * 8 + 4] = cc[s];
  g8[i * 8 + 5] = al[s];
  g8[i * 8 + 6] = 0.0f;
  g8[i * 8 + 7] = 0.0f;
}

// ---------------------------------------------------------------------------
// Phase 3b: build WMMA A-operand stream (16x4 f32 tiles of channel-padded
// sorted colors). ISA A layout: lane l holds A[M=l%16][K=2*(l/16)+j] in VGPR j.
// aBuf[chunk*64 + lane*2 + j].
// ---------------------------------------------------------------------------
__global__ void gs_abuf(const int* __restrict__ order, const float* __restrict__ col,
                        float* __restrict__ aBuf, int nElem) {
  int t = blockIdx.x * blockDim.x + threadIdx.x;
  if (t >= nElem) return;
  int chunk = t >> 6;
  int r = t & 63;
  int lane = r >> 1;
  int j = r & 1;
  int m = lane & 15;
  int k = ((lane >> 4) << 1) | j;
  int g = chunk * 4 + k;
  float v = 0.0f;
  if (m < 3) v = col[order[g] * 3 + m];
  aBuf[chunk * 64 + lane * 2 + j] = v;
}

// ---------------------------------------------------------------------------
// Phase 4: render. Each wave owns 32 contiguous pixels (two 16-px WMMA tiles).
// Lane l owns pixel base+l and runs the sequential transmittance scan.
// Color accumulation acc[3,16] += colors^T[3(pad 16),4] @ w[4,16] via
// V_WMMA_F32_16X16X4_F32 (f32 accumulate preserves reference precision).
// B layout (4x16): VGPR0 lanes0-15=K0 / lanes16-31=K2 ; VGPR1 = K1 / K3.
// ---------------------------------------------------------------------------
__global__ __launch_bounds__(256) void gs_render(const float* __restrict__ g8,
                                                 const float* __restrict__ aBuf,
                                                 float* __restrict__ out,
                                                 int nChunks, int W, int HW) {
  int tid = blockIdx.x * blockDim.x + threadIdx.x;
  int lane = tid & 31;
  int base = (tid >> 5) * 32;
  int pix = base + lane;
  float px = (float)(pix % W);
  float py = (float)(pix / W);

  float T = 1.0f;
  v8f CA = {0.0f, 0.0f, 0.0f, 0.0f, 0.0f, 0.0f, 0.0f, 0.0f};
  v8f CB = {0.0f, 0.0f, 0.0f, 0.0f, 0.0f, 0.0f, 0.0f, 0.0f};
  const v2f* aB = (const v2f*)aBuf;

  for (int ch = 0; ch < nChunks; ++ch) {
    v2f Av = aB[ch * 32 + lane];            // coalesced b64, per-chunk A operand
    const float* gp = g8 + ch * 32;         // uniform address -> scalar loads
    float warr[4];
#pragma unroll
    for (int k = 0; k < 4; ++k) {
      float gx = gp[k * 8 + 0], gy = gp[k * 8 + 1];
      float ga = gp[k * 8 + 2], gb = gp[k * 8 + 3];
      float gc = gp[k * 8 + 4], gal = gp[k * 8 + 5];
      float dx = px - gx, dy = py - gy;
      float pw = -0.5f * (ga * dx * dx + gc * dy * dy) - gb * dx * dy;
      float alp = fminf(0.99f, gal * __expf(pw));
      alp = (pw <= 0.0f && alp >= (1.0f / 255.0f)) ? alp : 0.0f;
      warr[k] = (T > 0.0001f) ? alp * T : 0.0f;
      T *= (1.0f - alp);
    }
    // half-wave swaps to build B operands for both 16-pixel tiles
    float s0 = __shfl_xor(warr[0], 16, 32);
    float s1 = __shfl_xor(warr[1], 16, 32);
    float s2 = __shfl_xor(warr[2], 16, 32);
    float s3 = __shfl_xor(warr[3], 16, 32);
    bool lo = (lane < 16);
    v2f BA, BB;
    BA.x = lo ? warr[0] : s2;  // tile A (pixels base..base+15)
    BA.y = lo ? warr[1] : s3;
    BB.x = lo ? s0 : warr[2];  // tile B (pixels base+16..base+31)
    BB.y = lo ? s1 : warr[3];
    CA = __builtin_amdgcn_wmma_f32_16x16x4_f32(false, Av, false, BA, (short)0, CA, false, false);
    CB = __builtin_amdgcn_wmma_f32_16x16x4_f32(false, Av, false, BB, (short)0, CB, false, false);
  }

  // D layout: VGPR r, lanes 0-15 hold row M=r, N=lane. Rows 0..2 = channels.
  if (lane < 16) {
    int pA = base + lane;
    int pB = base + 16 + lane;
    out[0 * HW + pA] = CA[0];
    out[1 * HW + pA] = CA[1];
    out[2 * HW + pA] = CA[2];
    out[0 * HW + pB] = CB[0];
    out[1 * HW + pB] = CB[1];
    out[2 * HW + pB] = CB[2];
  }
}

// ---------------------------------------------------------------------------
extern "C" void kernel_launch(void* const* d_in, const int* in_sizes, int n_in,
                              void* d_out, int out_size, void* d_ws, size_t ws_size,
                              hipStream_t stream) {
  const float* pws  = (const float*)d_in[0];
  const float* shs  = (const float*)d_in[1];
  const float* araw = (const float*)d_in[2];
  const float* sraw = (const float*)d_in[3];
  const float* rraw = (const float*)d_in[4];
  // d_in[5] = us (zeros, cancels in forward)
  const float* Rcw  = (const float*)d_in[6];
  const float* tcw  = (const float*)d_in[7];
  const float* intr = (const float*)d_in[8];
  // d_in[9]=height, d_in[10]=width: device scalars; dims fixed at 256 per setup

  const int N = in_sizes[0] / 3;  // 8192
  const int H = GS_H, W = GS_W, HW = GS_HW;
  const int nChunks = N / 4;

  float* out = (float*)d_out;
  float* areas = out + 3 * HW;

  // workspace layout (floats)
  float* ws    = (float*)d_ws;
  float* depth = ws;             // N
  float* ux    = ws + 1 * N;
  float* uy    = ws + 2 * N;
  float* ca    = ws + 3 * N;
  float* cb    = ws + 4 * N;
  float* cc    = ws + 5 * N;
  float* al    = ws + 6 * N;
  float* col   = ws + 7 * N;     // 3N
  int*   order = (int*)(ws + 10 * N);  // N ints
  float* g8    = ws + 11 * N;    // 8N
  float* aBuf  = ws + 19 * N;    // 16N

  gs_prep<<<(N + 255) / 256, 256, 0, stream>>>(pws, shs, araw, sraw, rraw, Rcw, tcw, intr,
                                               depth, ux, uy, ca, cb, cc, al, col, areas,
                                               N, H, W);
  gs_sort<<<1, 1024, 0, stream>>>(depth, order, N);
  gs_gather<<<(N + 255) / 256, 256, 0, stream>>>(order, ux, uy, ca, cb, cc, al, g8, N);
  int nA = nChunks * 64;
  gs_abuf<<<(nA + 255) / 256, 256, 0, stream>>>(order, col, aBuf, nA);
  gs_render<<<HW / 256, 256, 0, stream>>>(g8, aBuf, out, nChunks, W, HW);
}